// D2_S_atten_39831526703210
// MI455X (gfx1250) — compile-verified
//
#include <hip/hip_runtime.h>

// Problem constants (from reference setup_inputs)
#define B_ 32
#define C_ 64
#define H_ 64
#define W_ 64
#define HP 66   // padded height
#define WP 66   // padded width
#define NTILES 256   // (H*W)/16 spatial tiles per image

typedef __attribute__((ext_vector_type(2))) float v2f;
typedef __attribute__((ext_vector_type(8))) float v8f;

// Workspace layout, in floats. Total ~9.56M floats = ~38.3 MB.
static constexpr size_t WT_OFF  = 0;                       // wt[3][9][64][64] (q,k,v)
static constexpr size_t WT_SZ   = (size_t)3 * 9 * 64 * 64;
static constexpr size_t XT_OFF  = WT_OFF + WT_SZ;          // xt[B][66][66][64] zero-padded NHWC
static constexpr size_t XT_SZ   = (size_t)B_ * HP * WP * C_;
static constexpr size_t PART_OFF = XT_OFF + XT_SZ;         // partials[B][256][64]
static constexpr size_t PART_SZ  = (size_t)B_ * NTILES * C_;
static constexpr size_t KV_OFF   = PART_OFF + PART_SZ;     // kv[B][64]

// ---------------------------------------------------------------- kernel 0a
__global__ void zero_ws_kernel(float* __restrict__ p, size_t n) {
    size_t i = (size_t)blockIdx.x * blockDim.x + threadIdx.x;
    size_t stride = (size_t)gridDim.x * blockDim.x;
    for (; i < n; i += stride) p[i] = 0.0f;
}

// ---------------------------------------------------------------- kernel 0b
// wt[conv][tap][cin][cout] = w_conv[cout][cin][ty][tx]   (conv: 0=q,1=k,2=v)
__global__ void weight_transpose_kernel(const float* __restrict__ wq,
                                        const float* __restrict__ wk,
                                        const float* __restrict__ wv,
                                        float* __restrict__ wt) {
    int idx = blockIdx.x * blockDim.x + threadIdx.x;
    if (idx >= 3 * 9 * 64 * 64) return;
    int conv = idx / (9 * 64 * 64);
    int rem  = idx % (9 * 64 * 64);
    int tap  = rem / (64 * 64);
    int cin  = (rem / 64) & 63;
    int cout = rem & 63;
    const float* src = (conv == 0) ? wq : (conv == 1) ? wk : wv;
    wt[idx] = src[((size_t)cout * 64 + cin) * 9 + tap];
}

// ---------------------------------------------------------------- kernel 0c
// NCHW -> padded NHWC via LDS tile transpose (coalesced both sides).
__global__ __launch_bounds__(1024) void nhwc_pack_kernel(const float* __restrict__ x,
                                                         float* __restrict__ xt) {
    __shared__ float tile[32][33];
    int w0 = blockIdx.x * 32;       // 0 or 32
    int c0 = blockIdx.y * 32;       // 0 or 32
    int bh = blockIdx.z;            // b*H + h
    int b = bh / H_, h = bh % H_;
    // read: coalesced along w
    int w = w0 + threadIdx.x;
    int c = c0 + threadIdx.y;
    tile[threadIdx.y][threadIdx.x] =
        x[(((size_t)b * C_ + c) * H_ + h) * W_ + w];
    __syncthreads();
    // write: coalesced along c
    int cw = c0 + threadIdx.x;
    int ww = w0 + threadIdx.y;
    xt[(((size_t)b * HP + (h + 1)) * WP + (ww + 1)) * C_ + cw] =
        tile[threadIdx.x][threadIdx.y];
}

// ---------------------------------------------------------------- kernel 1
// Per wave: 16 spatial positions (one row segment). Compute k- and v-conv
// tiles with V_WMMA_F32_16X16X4_F32, then partial kv = sum_pos (k+bk)*(v+bv).
__global__ __launch_bounds__(32) void conv_kv_kernel(const float* __restrict__ xt,
                                                     const float* __restrict__ wt,
                                                     const float* __restrict__ bk,
                                                     const float* __restrict__ bv,
                                                     float* __restrict__ partials) {
    const int lane  = threadIdx.x;        // 0..31
    const int wtile = blockIdx.x;         // 0..3
    const int h     = blockIdx.y;         // 0..63
    const int b     = blockIdx.z;         // 0..31
    const int w0 = wtile * 16;
    const int m  = lane & 15;             // A: M = lane%16 ; B/D: N = lane%16
    const int kh = (lane >> 4) * 2;       // K half: lanes>=16 hold K=2,3

    const float* wtk = wt + (size_t)1 * 9 * 64 * 64;
    const float* wtv = wt + (size_t)2 * 9 * 64 * 64;

    v8f accK[4], accV[4];
#pragma unroll
    for (int s = 0; s < 4; ++s) {
        accK[s] = (v8f){0.f, 0.f, 0.f, 0.f, 0.f, 0.f, 0.f, 0.f};
        accV[s] = (v8f){0.f, 0.f, 0.f, 0.f, 0.f, 0.f, 0.f, 0.f};
    }

    for (int tap = 0; tap < 9; ++tap) {
        const int dy = tap / 3, dx = tap % 3;
        const float* abase =
            xt + (((size_t)(b * HP + h + dy)) * WP + (w0 + m + dx)) * C_ + kh;
        const float* wkb = wtk + (size_t)tap * 64 * 64;
        const float* wvb = wtv + (size_t)tap * 64 * 64;
        for (int c = 0; c < 64; c += 4) {
            v2f a = *(const v2f*)(abase + c);   // contiguous cin pair (NHWC)
            const int row = c + kh;
#pragma unroll
            for (int s = 0; s < 4; ++s) {
                v2f bkf, bvf;
                bkf.x = wkb[(size_t)row * 64 + s * 16 + m];
                bkf.y = wkb[(size_t)(row + 1) * 64 + s * 16 + m];
                bvf.x = wvb[(size_t)row * 64 + s * 16 + m];
                bvf.y = wvb[(size_t)(row + 1) * 64 + s * 16 + m];
                accK[s] = __builtin_amdgcn_wmma_f32_16x16x4_f32(
                    false, a, false, bkf, (short)0, accK[s], false, false);
                accV[s] = __builtin_amdgcn_wmma_f32_16x16x4_f32(
                    false, a, false, bvf, (short)0, accV[s], false, false);
            }
        }
    }

    // Reduce: each lane holds 8 positions for cout = s*16 + m.
    const size_t pbase = ((size_t)b * NTILES + (h * 4 + wtile)) * C_;
#pragma unroll
    for (int s = 0; s < 4; ++s) {
        const float bkc = bk[s * 16 + m];
        const float bvc = bv[s * 16 + m];
        float p = 0.f;
#pragma unroll
        for (int g = 0; g < 8; ++g)
            p += (accK[s][g] + bkc) * (accV[s][g] + bvc);
        p += __shfl_xor(p, 16, 32);       // fold positions 8..15 into 0..7 half
        if (lane < 16) partials[pbase + s * 16 + lane] = p;
    }
}

// ---------------------------------------------------------------- kernel 2
__global__ void kv_reduce_kernel(const float* __restrict__ partials,
                                 float* __restrict__ kv) {
    int idx = blockIdx.x * blockDim.x + threadIdx.x;
    if (idx >= B_ * C_) return;
    int b = idx / C_, c = idx % C_;
    float s = 0.f;
    for (int t = 0; t < NTILES; ++t)
        s += partials[((size_t)b * NTILES + t) * C_ + c];
    kv[idx] = s;
}

// ---------------------------------------------------------------- kernel 3
// q-conv via WMMA, then out = gamma*(q+bq)*kv[b,c] + x  (NCHW writes).
__global__ __launch_bounds__(32) void conv_q_out_kernel(const float* __restrict__ xt,
                                                        const float* __restrict__ wt,
                                                        const float* __restrict__ bq,
                                                        const float* __restrict__ kv,
                                                        const float* __restrict__ x,
                                                        const float* __restrict__ gamma,
                                                        float* __restrict__ out) {
    const int lane  = threadIdx.x;
    const int wtile = blockIdx.x;
    const int h     = blockIdx.y;
    const int b     = blockIdx.z;
    const int w0 = wtile * 16;
    const int m  = lane & 15;
    const int kh = (lane >> 4) * 2;

    v8f accQ[4];
#pragma unroll
    for (int s = 0; s < 4; ++s)
        accQ[s] = (v8f){0.f, 0.f, 0.f, 0.f, 0.f, 0.f, 0.f, 0.f};

    for (int tap = 0; tap < 9; ++tap) {
        const int dy = tap / 3, dx = tap % 3;
        const float* abase =
            xt + (((size_t)(b * HP + h + dy)) * WP + (w0 + m + dx)) * C_ + kh;
        const float* wqb = wt + (size_t)tap * 64 * 64;   // conv 0 = q
        for (int c = 0; c < 64; c += 4) {
            v2f a = *(const v2f*)(abase + c);
            const int row = c + kh;
#pragma unroll
            for (int s = 0; s < 4; ++s) {
                v2f bf;
                bf.x = wqb[(size_t)row * 64 + s * 16 + m];
                bf.y = wqb[(size_t)(row + 1) * 64 + s * 16 + m];
                accQ[s] = __builtin_amdgcn_wmma_f32_16x16x4_f32(
                    false, a, false, bf, (short)0, accQ[s], false, false);
            }
        }
    }

    const float g0 = gamma[0];
    const int wbase = w0 + ((lane >> 4) * 8);   // D: lanes>=16 hold M=g+8
#pragma unroll
    for (int s = 0; s < 4; ++s) {
        const int cout = s * 16 + m;
        const float bqc = bq[cout];
        const float kvc = kv[b * C_ + cout];
        const size_t rowoff = (((size_t)(b * C_ + cout)) * H_ + h) * W_;
#pragma unroll
        for (int g = 0; g < 8; ++g) {
            const size_t idx = rowoff + wbase + g;
            out[idx] = g0 * (accQ[s][g] + bqc) * kvc + x[idx];
        }
    }
}

// ---------------------------------------------------------------- launch
extern "C" void kernel_launch(void* const* d_in, const int* in_sizes, int n_in,
                              void* d_out, int out_size, void* d_ws, size_t ws_size,
                              hipStream_t stream) {
    const float* x     = (const float*)d_in[0];
    const float* wq    = (const float*)d_in[1];
    const float* bq    = (const float*)d_in[2];
    const float* wk    = (const float*)d_in[3];
    const float* bk    = (const float*)d_in[4];
    const float* wv    = (const float*)d_in[5];
    const float* bv    = (const float*)d_in[6];
    const float* gamma = (const float*)d_in[7];
    float* out = (float*)d_out;

    float* ws   = (float*)d_ws;          // needs ~38.3 MB
    float* wt   = ws + WT_OFF;
    float* xt   = ws + XT_OFF;
    float* part = ws + PART_OFF;
    float* kv   = ws + KV_OFF;

    // 0a: zero the padded NHWC buffer (border zeros = SAME padding)
    zero_ws_kernel<<<4096, 256, 0, stream>>>(xt, XT_SZ);
    // 0b: transpose weights to wt[conv][tap][cin][cout]
    weight_transpose_kernel<<<(3 * 9 * 64 * 64 + 255) / 256, 256, 0, stream>>>(wq, wk, wv, wt);
    // 0c: pack x into padded NHWC
    {
        dim3 grid(W_ / 32, C_ / 32, B_ * H_);
        dim3 block(32, 32, 1);
        nhwc_pack_kernel<<<grid, block, 0, stream>>>(x, xt);
    }
    // 1: k,v conv tiles (WMMA) + per-tile kv partials
    {
        dim3 grid(4, H_, B_);
        conv_kv_kernel<<<grid, 32, 0, stream>>>(xt, wt, bk, bv, part);
    }
    // 2: deterministic reduction of partials -> kv[B][C]
    kv_reduce_kernel<<<(B_ * C_ + 255) / 256, 256, 0, stream>>>(part, kv);
    // 3: q conv (WMMA) + epilogue out = gamma*q*kv + x
    {
        dim3 grid(4, H_, B_);
        conv_q_out_kernel<<<grid, 32, 0, stream>>>(xt, wt, bq, kv, x, gamma, out);
    }
}